// CustomizedRelaModule_798863917374
// MI455X (gfx1250) — compile-verified
//
#include <hip/hip_runtime.h>

// Problem constants (from reference): N=4096 samples, V=256 channels, NH=64.
#define NS   4096
#define VC   256
#define NH   64
#define KD   256          // GEMM K == V
#define ROWS 128          // rows per block (8 waves x 16)
#define NTHREADS 256

// LDS holds W_c pre-paired along K:  pair row kp = (W[2kp][col], W[2kp+1][col]).
// Even-kp pairs (k = 4r, 4r+1) in ldsE, odd-kp pairs (k = 4r+2, 4r+3) in ldsO.
// ldsO is offset from ldsE by 16 extra float2 (128 B) -> half-waves hit bank
// sets 32 apart: conflict-free ds_load_b64, and both halves share row index k/4.
#define PAIR_ROWS  64                 // 64 pair-rows per array (k/4 = 0..63)
#define ARR_ELEMS  (PAIR_ROWS * NH)   // 4096 float2 per array
#define GAP_ELEMS  16                 // 128-byte gap => 32-bank shift

typedef __attribute__((ext_vector_type(2))) float v2f;
typedef __attribute__((ext_vector_type(8))) float v8f;

__device__ __forceinline__ float hw_tanh(float x) {
#if __has_builtin(__builtin_amdgcn_tanhf)
    return __builtin_amdgcn_tanhf(x);      // gfx1250 V_TANH_F32 (trans pipe)
#else
    return tanhf(x);
#endif
}

__global__ __launch_bounds__(NTHREADS)
void rela_fused_wmma(const float* __restrict__ data,
                     const float* __restrict__ adj,
                     const float* __restrict__ neurons,
                     const float* __restrict__ w_in,
                     const float* __restrict__ b_in,
                     const float* __restrict__ w_out,
                     const float* __restrict__ b_out,
                     float* __restrict__ out)   // (N, V) region of d_out
{
    __shared__ v2f ldsW[2 * ARR_ELEMS + GAP_ELEMS];
    v2f* const ldsE = ldsW;
    v2f* const ldsO = ldsW + ARR_ELEMS + GAP_ELEMS;

    const int c   = blockIdx.y;
    const int tid = threadIdx.x;

    // ---- Build W_c in LDS: W[j][h] = (j==c) ? 0 : adj[j,c] * w_in[c, j-(j>c), h]
    // perm (67MB) is a fixed drop-diagonal selector; never read from memory.
    for (int idx = tid; idx < ARR_ELEMS; idx += NTHREADS) {
        const int r   = idx >> 6;          // pair-row group: covers j = 4r .. 4r+3
        const int col = idx & 63;          // NH column (consecutive tids -> coalesced)
        float w[4];
        #pragma unroll
        for (int q = 0; q < 4; ++q) {
            const int j = 4 * r + q;
            float v = 0.0f;
            if (j != c) {
                const int kidx = j - (j > c ? 1 : 0);
                v = adj[(size_t)j * VC + c] *
                    w_in[((size_t)c * (VC - 1) + kidx) * NH + col];
            }
            w[q] = v;
        }
        ldsE[idx] = (v2f){w[0], w[1]};     // k = 4r, 4r+1   (even kp)
        ldsO[idx] = (v2f){w[2], w[3]};     // k = 4r+2, 4r+3 (odd kp)
    }
    __syncthreads();

    const int wave = tid >> 5;
    const int lane = tid & 31;
    const int half = lane >> 4;            // 0: K={0,1}, 1: K={2,3} per ISA A/B layout
    const int l16  = lane & 15;

    const int r0 = blockIdx.x * ROWS + wave * 16;
    const float* __restrict__ arow = data + (size_t)(r0 + l16) * KD + half * 2;

    // One LDS address per N-tile, fixed for the whole K-loop; per-step row is an
    // immediate DS offset (k/4 * 64 float2 = k*128 bytes, max 32256).
    const v2f* const bbase = half ? ldsO : ldsE;
    const v2f* const b0p = bbase + ( 0 + l16);
    const v2f* const b1p = bbase + (16 + l16);
    const v2f* const b2p = bbase + (32 + l16);
    const v2f* const b3p = bbase + (48 + l16);

    v8f acc0 = {}, acc1 = {}, acc2 = {}, acc3 = {};

    #pragma unroll
    for (int k = 0; k < KD; k += 4) {
        if ((k & 15) == 0 && (k + 16) < KD)
            __builtin_prefetch(arow + k + 16, 0, 3);   // near-cache global_prefetch_b8

        // A fragment (16x4 f32): lane L holds data[r0+(L&15)][k + 2*half .. +1]
        const v2f a = *(const v2f*)(arow + k);

        const int pr = (k >> 2) * NH;       // pair-row offset (compile-time per step)
        const v2f b0 = b0p[pr];
        const v2f b1 = b1p[pr];
        const v2f b2 = b2p[pr];
        const v2f b3 = b3p[pr];

        acc0 = __builtin_amdgcn_wmma_f32_16x16x4_f32(false, a, false, b0, (short)0, acc0, false, false);
        acc1 = __builtin_amdgcn_wmma_f32_16x16x4_f32(false, a, false, b1, (short)0, acc1, false, false);
        acc2 = __builtin_amdgcn_wmma_f32_16x16x4_f32(false, a, false, b2, (short)0, acc2, false, false);
        acc3 = __builtin_amdgcn_wmma_f32_16x16x4_f32(false, a, false, b3, (short)0, acc3, false, false);
    }

    // ---- Fused epilogue: tanh, gate by neurons^T * w_out, reduce over NH, +b_out
    float bin[4], g[4];
    #pragma unroll
    for (int t = 0; t < 4; ++t) {
        const int col = t * 16 + l16;                     // NH column this lane owns
        bin[t] = b_in[(size_t)c * NH + col];
        g[t]   = neurons[(size_t)col * VC + c] * w_out[(size_t)c * NH + col];
    }
    const float bo = b_out[c];

    // C/D layout: VGPR v holds row r0 + v + 8*half at column t*16 + l16
    #pragma unroll
    for (int v = 0; v < 8; ++v) {
        float s = hw_tanh(acc0[v] + bin[0]) * g[0]
                + hw_tanh(acc1[v] + bin[1]) * g[1]
                + hw_tanh(acc2[v] + bin[2]) * g[2]
                + hw_tanh(acc3[v] + bin[3]) * g[3];
        // reduce the 16 NH-columns held by each half-wave (xor of bits 0..3 stays in-half)
        s += __shfl_xor(s, 1, 32);
        s += __shfl_xor(s, 2, 32);
        s += __shfl_xor(s, 4, 32);
        s += __shfl_xor(s, 8, 32);
        if (l16 == 0) {
            const int row = r0 + v + half * 8;
            out[(size_t)row * VC + c] = s + bo;
        }
    }
}

extern "C" void kernel_launch(void* const* d_in, const int* in_sizes, int n_in,
                              void* d_out, int out_size, void* d_ws, size_t ws_size,
                              hipStream_t stream) {
    const float* data    = (const float*)d_in[0];
    const float* adj     = (const float*)d_in[1];
    const float* neurons = (const float*)d_in[2];
    const float* w_in    = (const float*)d_in[3];
    const float* b_in    = (const float*)d_in[4];
    const float* w_out   = (const float*)d_in[5];
    const float* b_out   = (const float*)d_in[6];
    // d_in[7] (perm) intentionally unused.

    float* outp = (float*)d_out;

    // Output tuple: (adj, out) concatenated flat.
    hipMemcpyAsync(outp, adj, (size_t)VC * VC * sizeof(float),
                   hipMemcpyDeviceToDevice, stream);

    dim3 grid(NS / ROWS, VC);   // (32, 256)
    rela_fused_wmma<<<grid, NTHREADS, 0, stream>>>(
        data, adj, neurons, w_in, b_in, w_out, b_out, outp + (size_t)VC * VC);
}